// AlignmentLayer_10531259810179
// MI455X (gfx1250) — compile-verified
//
#include <hip/hip_runtime.h>

typedef float v2f __attribute__((ext_vector_type(2)));
typedef float v8f __attribute__((ext_vector_type(8)));

// ---------------------------------------------------------------------------
// Kernel 0: center the reference structure.  ref0[n] = ref_x[n] - mean(ref_x)
// ---------------------------------------------------------------------------
__global__ void k_center_ref(const float* __restrict__ ref_x,
                             float* __restrict__ ref0, int NR) {
    __shared__ float mean[3];
    int t = threadIdx.x;
    if (t < 3) {
        float s = 0.0f;
        for (int n = 0; n < NR; ++n) s += ref_x[n * 3 + t];
        mean[t] = s / (float)NR;
    }
    __syncthreads();
    for (int i = t; i < NR * 3; i += blockDim.x)
        ref0[i] = ref_x[i] - mean[i % 3];
}

// ---------------------------------------------------------------------------
// Jacobi rotation on symmetric 3x3 K, accumulating eigenvectors in V.
// Compile-time indices keep everything in registers (SROA-friendly).
// ---------------------------------------------------------------------------
template <int P_, int Q_, int R_>
__device__ __forceinline__ void jacobi_rot(float K[3][3], float V[3][3]) {
    float apq = K[P_][Q_];
    if (fabsf(apq) > 1e-20f) {
        float app = K[P_][P_], aqq = K[Q_][Q_];
        float tau = (aqq - app) / (2.0f * apq);
        float t   = copysignf(1.0f, tau) / (fabsf(tau) + sqrtf(1.0f + tau * tau));
        float c   = 1.0f / sqrtf(1.0f + t * t);
        float s   = t * c;
        K[P_][P_] = app - t * apq;
        K[Q_][Q_] = aqq + t * apq;
        K[P_][Q_] = 0.0f; K[Q_][P_] = 0.0f;
        float krp = K[R_][P_], krq = K[R_][Q_];
        K[R_][P_] = K[P_][R_] = c * krp - s * krq;
        K[R_][Q_] = K[Q_][R_] = s * krp + c * krq;
#pragma unroll
        for (int i = 0; i < 3; ++i) {
            float vip = V[i][P_], viq = V[i][Q_];
            V[i][P_] = c * vip - s * viq;
            V[i][Q_] = s * vip + c * viq;
        }
    }
}

// ---------------------------------------------------------------------------
// Kernel A: one thread per frame.  Gather 64 atoms, covariance P, Kabsch
// rotation via Jacobi eigensolve of P^T P.  Writes [xc(3), R^T(9)] per frame.
// Note: ref0 columns sum to zero, so P needs no centering of the gathered set.
// ---------------------------------------------------------------------------
__global__ void k_frame_rot(const float* __restrict__ x,
                            const float* __restrict__ ref0,
                            const int* __restrict__ idx,
                            float* __restrict__ fws,
                            int L, int N, int NR) {
    int l = blockIdx.x * blockDim.x + threadIdx.x;
    if (l >= L) return;
    const float* xl = x + (size_t)l * N * 3;

    float sx = 0.f, sy = 0.f, sz = 0.f;
    float P[3][3] = {};
    for (int n = 0; n < NR; ++n) {
        int a = idx[n];
        float px = xl[a * 3 + 0], py = xl[a * 3 + 1], pz = xl[a * 3 + 2];
        float r0 = ref0[n * 3 + 0], r1 = ref0[n * 3 + 1], r2 = ref0[n * 3 + 2];
        sx += px; sy += py; sz += pz;
        P[0][0] += px * r0; P[0][1] += px * r1; P[0][2] += px * r2;
        P[1][0] += py * r0; P[1][1] += py * r1; P[1][2] += py * r2;
        P[2][0] += pz * r0; P[2][1] += pz * r1; P[2][2] += pz * r2;
    }
    float inv = 1.0f / (float)NR;
    float xc0 = sx * inv, xc1 = sy * inv, xc2 = sz * inv;

    // K = P^T P  (symmetric PSD), eigensolve by cyclic Jacobi
    float K[3][3], V[3][3] = {{1, 0, 0}, {0, 1, 0}, {0, 0, 1}};
#pragma unroll
    for (int i = 0; i < 3; ++i)
#pragma unroll
        for (int j = 0; j < 3; ++j)
            K[i][j] = P[0][i] * P[0][j] + P[1][i] * P[1][j] + P[2][i] * P[2][j];

    for (int sweep = 0; sweep < 10; ++sweep) {
        jacobi_rot<0, 1, 2>(K, V);
        jacobi_rot<0, 2, 1>(K, V);
        jacobi_rot<1, 2, 0>(K, V);
    }

    float lam[3] = {K[0][0], K[1][1], K[2][2]};
    // sort eigenpairs descending (swap lam and V columns)
#define CSWAP(i, j)                                                        \
    if (lam[i] < lam[j]) {                                                 \
        float tl = lam[i]; lam[i] = lam[j]; lam[j] = tl;                   \
        float t0 = V[0][i]; V[0][i] = V[0][j]; V[0][j] = t0;               \
        float t1 = V[1][i]; V[1][i] = V[1][j]; V[1][j] = t1;               \
        float t2 = V[2][i]; V[2][i] = V[2][j]; V[2][j] = t2;               \
    }
    CSWAP(0, 1); CSWAP(1, 2); CSWAP(0, 1);
#undef CSWAP

    float sig0 = sqrtf(fmaxf(lam[0], 0.0f));
    float sig2 = sqrtf(fmaxf(lam[2], 0.0f));

    float U[3][3];  // U[a][j] = component a of left singular vector u_j
#pragma unroll
    for (int j = 0; j < 2; ++j) {
        float w0 = P[0][0] * V[0][j] + P[0][1] * V[1][j] + P[0][2] * V[2][j];
        float w1 = P[1][0] * V[0][j] + P[1][1] * V[1][j] + P[1][2] * V[2][j];
        float w2 = P[2][0] * V[0][j] + P[2][1] * V[1][j] + P[2][2] * V[2][j];
        float nrm = sqrtf(w0 * w0 + w1 * w1 + w2 * w2);
        float ninv = (nrm > 1e-20f) ? (1.0f / nrm) : 0.0f;
        U[0][j] = w0 * ninv; U[1][j] = w1 * ninv; U[2][j] = w2 * ninv;
    }
    {
        float w0 = P[0][0] * V[0][2] + P[0][1] * V[1][2] + P[0][2] * V[2][2];
        float w1 = P[1][0] * V[0][2] + P[1][1] * V[1][2] + P[1][2] * V[2][2];
        float w2 = P[2][0] * V[0][2] + P[2][1] * V[1][2] + P[2][2] * V[2][2];
        float nrm = sqrtf(w0 * w0 + w1 * w1 + w2 * w2);
        if (sig2 > 1e-6f * sig0 && nrm > 1e-20f) {
            float ninv = 1.0f / nrm;
            U[0][2] = w0 * ninv; U[1][2] = w1 * ninv; U[2][2] = w2 * ninv;
        } else {  // degenerate: u2 = u0 x u1
            U[0][2] = U[1][0] * U[2][1] - U[2][0] * U[1][1];
            U[1][2] = U[2][0] * U[0][1] - U[0][0] * U[2][1];
            U[2][2] = U[0][0] * U[1][1] - U[1][0] * U[0][1];
        }
    }

    // sign(det(U V^T)) == sign(det(P)) since singular values >= 0
    float detP = P[0][0] * (P[1][1] * P[2][2] - P[1][2] * P[2][1])
               - P[0][1] * (P[1][0] * P[2][2] - P[1][2] * P[2][0])
               + P[0][2] * (P[1][0] * P[2][1] - P[1][1] * P[2][0]);
    float d2 = (detP < 0.0f) ? -1.0f : 1.0f;

    // R[a][b] = sum_j d_j * U[a][j] * V[b][j];  store RT[m][k] = R[k][m]
    float* o = fws + (size_t)l * 12;
    o[0] = xc0; o[1] = xc1; o[2] = xc2;
#pragma unroll
    for (int m = 0; m < 3; ++m)
#pragma unroll
        for (int k = 0; k < 3; ++k)
            o[3 + m * 3 + k] =
                U[k][0] * V[m][0] + U[k][1] * V[m][1] + d2 * U[k][2] * V[m][2];
}

// ---------------------------------------------------------------------------
// Kernel B: out = (x - xc) @ R, via v_wmma_f32_16x16x4_f32 computing
// out^T = R^T * xn^T.  One wave handles 16-atom tiles (tile stride = 8 waves).
//   A (16x4, 2 VGPRs): lanes 0-15 hold {K0,K1}, lanes 16-31 hold {K2,K3};
//                       A[m][k] = R^T[m][k] for m,k < 3, else 0.
//   B (4x16, 2 VGPRs): v0 = rows K0|K1, v1 = rows K2|K3 -> xn coords x|y, z|0.
//   D rows 0..2 (lanes 0-15, c[0..2]) = float3 output of atom `lane`.
// ---------------------------------------------------------------------------
__global__ void k_apply(const float* __restrict__ x,
                        const float* __restrict__ fws,
                        float* __restrict__ out, int N) {
    int l    = blockIdx.x;
    int lane = threadIdx.x & 31;
    int wv   = threadIdx.x >> 5;     // 0..7 (blockDim = 256, wave32)
    int m    = lane & 15;
    bool hi  = lane >= 16;

    const float* fw = fws + (size_t)l * 12;
    float xc0 = fw[0], xc1 = fw[1], xc2 = fw[2];
    const float* RT = fw + 3;

    v2f a; a.x = 0.0f; a.y = 0.0f;
    if (m < 3) {
        a.x = hi ? RT[m * 3 + 2] : RT[m * 3 + 0];
        a.y = hi ? 0.0f          : RT[m * 3 + 1];
    }

    const float* xl = x   + (size_t)l * N * 3;
    float*       ol = out + (size_t)l * N * 3;
    int tiles = (N + 15) >> 4;

    for (int tile = wv; tile < tiles; tile += 8) {
        int atom = (tile << 4) + m;
        v2f b; b.x = 0.0f; b.y = 0.0f;
        if (atom < N) {
            const float* p = xl + atom * 3;
            float v0 = p[hi ? 1 : 0];            // x (lo lanes) | y (hi lanes)
            float v1 = p[2];                     // z
            b.x = v0 - (hi ? xc1 : xc0);
            b.y = hi ? 0.0f : (v1 - xc2);
        }
        v8f c = {};
        c = __builtin_amdgcn_wmma_f32_16x16x4_f32(
                /*neg_a=*/false, a, /*neg_b=*/false, b,
                /*c_mod=*/(short)0, c, /*reuse_a=*/false, /*reuse_b=*/false);
        if (!hi && atom < N) {
            float* q = ol + atom * 3;
            q[0] = c[0]; q[1] = c[1]; q[2] = c[2];
        }
    }
}

// ---------------------------------------------------------------------------
extern "C" void kernel_launch(void* const* d_in, const int* in_sizes, int n_in,
                              void* d_out, int out_size, void* d_ws, size_t ws_size,
                              hipStream_t stream) {
    const float* x     = (const float*)d_in[0];
    const float* ref_x = (const float*)d_in[1];
    const int*   idx   = (const int*)d_in[2];
    float*       out   = (float*)d_out;
    float*       ws    = (float*)d_ws;

    const int NR = in_sizes[2];              // 64
    const int N  = 2000;                     // per reference setup_inputs
    const int L  = in_sizes[0] / (N * 3);    // 8192

    float* ref0 = ws;                        // NR*3 floats
    float* fws  = ws + NR * 3;               // L*12 floats: [xc(3), R^T(9)]

    k_center_ref<<<1, 192, 0, stream>>>(ref_x, ref0, NR);
    k_frame_rot<<<(L + 255) / 256, 256, 0, stream>>>(x, ref0, idx, fws, L, N, NR);
    k_apply<<<L, 256, 0, stream>>>(x, fws, out, N);
}